// LightGCN_5153960755404
// MI455X (gfx1250) — compile-verified
//
#include <hip/hip_runtime.h>
#include <math.h>

#define EMB 64
#define EPS 1e-12f

typedef __attribute__((ext_vector_type(2))) float v2f;
typedef __attribute__((ext_vector_type(8))) float v8f;

// cur = emb; acc = emb; next = 0   (float4 over n*16 vec4 elements)
__global__ void k_init(const float* __restrict__ emb, float* __restrict__ cur,
                       float* __restrict__ acc, float* __restrict__ nxt, long n4) {
    long i = (long)blockIdx.x * blockDim.x + threadIdx.x;
    if (i >= n4) return;
    float4 v = ((const float4*)emb)[i];
    ((float4*)cur)[i] = v;
    ((float4*)acc)[i] = v;
    ((float4*)nxt)[i] = make_float4(0.f, 0.f, 0.f, 0.f);
}

// COO SpMM: next[row] += val * cur[col].  16 lanes per edge, float4 per lane:
// a half-wave covers one contiguous 256B embedding row (coalesced b128 gather,
// contiguous f32 atomic adds resolved in L2 — whole working set fits in 192MB L2).
__global__ void k_spmm(const int* __restrict__ rows, const int* __restrict__ cols,
                       const float* __restrict__ vals, const float* __restrict__ cur,
                       float* __restrict__ nxt, int n_edges) {
    long tid = (long)blockIdx.x * blockDim.x + threadIdx.x;
    int e = (int)(tid >> 4);
    if (e >= n_edges) return;
    int seg = (int)(tid & 15);
    int r = rows[e];
    int c = cols[e];
    float w = vals[e];
    float4 x = *((const float4*)(cur + (long)c * EMB) + seg);
    float* dst = nxt + (long)r * EMB + seg * 4;
    atomicAdd(dst + 0, w * x.x);
    atomicAdd(dst + 1, w * x.y);
    atomicAdd(dst + 2, w * x.z);
    atomicAdd(dst + 3, w * x.w);
}

// acc += next; cur = next; next = 0   (float4)
__global__ void k_accum(float* __restrict__ acc, float* __restrict__ nxt,
                        float* __restrict__ cur, long n4) {
    long i = (long)blockIdx.x * blockDim.x + threadIdx.x;
    if (i >= n4) return;
    float4 v = ((float4*)nxt)[i];
    float4 a = ((float4*)acc)[i];
    a.x += v.x; a.y += v.y; a.z += v.z; a.w += v.w;
    ((float4*)acc)[i] = a;
    ((float4*)cur)[i] = v;
    ((float4*)nxt)[i] = make_float4(0.f, 0.f, 0.f, 0.f);
}

// Row-wise mean + L2 normalize using V_WMMA_F32_16X16X4_F32 for the 64-wide
// sum-of-squares reduction. One wave32 per 16 rows.
// A (16x4 f32): lanes 0-15 hold K=0,1 ; lanes 16-31 hold K=2,3 (per ISA layout)
//   -> lane loads float2 at dim = 4k + (lane>>4)*2 of row (lane&15), squared.
// B (4x16 f32): all ones (layout-independent).
// D (16x16 f32): every column n of row m equals sumsq(row m); lanes 0-15 hold
// rows 0-7 in d[0..7], lanes 16-31 hold rows 8-15.
__global__ void __launch_bounds__(32)
k_finalize_wmma(const float* __restrict__ acc, float* __restrict__ out, int nrows) {
    int row0 = blockIdx.x * 16;
    if (row0 + 16 > nrows) return;            // uniform; keeps EXEC all-1s for WMMA
    int lane = threadIdx.x;
    int m    = lane & 15;
    int half = lane >> 4;

    v8f c = {0.f, 0.f, 0.f, 0.f, 0.f, 0.f, 0.f, 0.f};
    v2f b; b[0] = 1.0f; b[1] = 1.0f;

    const float* rp = acc + (long)(row0 + m) * EMB + half * 2;
#pragma unroll
    for (int k = 0; k < 16; ++k) {
        float2 xv = *(const float2*)(rp + k * 4);
        v2f a; a[0] = xv.x * xv.x; a[1] = xv.y * xv.y;
        c = __builtin_amdgcn_wmma_f32_16x16x4_f32(
                /*neg_a=*/false, a, /*neg_b=*/false, b,
                /*c_mod=*/(short)0, c, /*reuse_a=*/false, /*reuse_b=*/false);
    }

    __shared__ float scale[16];
    if (m == 0) {
        int base = half * 8;
#pragma unroll
        for (int r = 0; r < 8; ++r) {
            float ss = c[r];
            // out = (acc/4) / max(||acc/4||, eps)
            float nrm = 0.25f * sqrtf(ss);
            scale[base + r] = 0.25f / fmaxf(nrm, EPS);
        }
    }
    __syncthreads();

#pragma unroll
    for (int r = 0; r < 16; ++r) {
        float s = scale[r];
        long off = (long)(row0 + r) * EMB + lane * 2;
        float2 xv = *(const float2*)(acc + off);
        float2 ov; ov.x = xv.x * s; ov.y = xv.y * s;
        *(float2*)(out + off) = ov;
    }
}

// Generic remainder path (shuffle reduction); unused for 100000/50000 rows.
__global__ void __launch_bounds__(32)
k_finalize_rem(const float* __restrict__ acc, float* __restrict__ out,
               int row_start, int nrows) {
    int row = row_start + blockIdx.x;
    if (row >= nrows) return;
    int lane = threadIdx.x;
    long off = (long)row * EMB + lane * 2;
    float2 x = *(const float2*)(acc + off);
    float ss = x.x * x.x + x.y * x.y;
#pragma unroll
    for (int msk = 1; msk < 32; msk <<= 1) ss += __shfl_xor(ss, msk, 32);
    float s = 0.25f / fmaxf(0.25f * sqrtf(ss), EPS);
    float2 o; o.x = x.x * s; o.y = x.y * s;
    *(float2*)(out + off) = o;
}

extern "C" void kernel_launch(void* const* d_in, const int* in_sizes, int n_in,
                              void* d_out, int out_size, void* d_ws, size_t ws_size,
                              hipStream_t stream) {
    const int*   m_rows = (const int*)d_in[0];
    const int*   m_cols = (const int*)d_in[1];
    const float* m_vals = (const float*)d_in[2];
    const int*   a_rows = (const int*)d_in[3];
    const int*   a_cols = (const int*)d_in[4];
    const float* a_vals = (const float*)d_in[5];
    const float* m_emb  = (const float*)d_in[6];
    const float* a_emb  = (const float*)d_in[7];

    int ne_m = in_sizes[0];
    int ne_a = in_sizes[3];
    int n_m  = in_sizes[6] / EMB;
    int n_a  = in_sizes[7] / EMB;

    float* out   = (float*)d_out;
    float* acc_m = out;                       // mashup output region doubles as acc
    float* acc_a = out + (long)n_m * EMB;     // api output region doubles as acc

    // Scratch ping-pong buffers sized for the larger graph, reused sequentially.
    float* bufCur = (float*)d_ws;
    float* bufNxt = bufCur + (long)n_m * EMB;

    const int B = 256;
    auto run = [&](const int* rows, const int* cols, const float* vals,
                   const float* emb, float* acc, int n, int ne) {
        long n4 = (long)n * (EMB / 4);
        long ginit = (n4 + B - 1) / B;
        k_init<<<(unsigned)ginit, B, 0, stream>>>(emb, bufCur, acc, bufNxt, n4);
        long tspmm = (long)ne * 16;
        long gspmm = (tspmm + B - 1) / B;
        for (int l = 0; l < 3; ++l) {
            k_spmm<<<(unsigned)gspmm, B, 0, stream>>>(rows, cols, vals, bufCur, bufNxt, ne);
            k_accum<<<(unsigned)ginit, B, 0, stream>>>(acc, bufNxt, bufCur, n4);
        }
        int ntiles = n / 16;
        if (ntiles > 0)
            k_finalize_wmma<<<ntiles, 32, 0, stream>>>(acc, acc, n);
        int rem = n - ntiles * 16;
        if (rem > 0)
            k_finalize_rem<<<rem, 32, 0, stream>>>(acc, acc, ntiles * 16, n);
    };

    run(m_rows, m_cols, m_vals, m_emb, acc_m, n_m, ne_m);
    run(a_rows, a_cols, a_vals, a_emb, acc_a, n_a, ne_a);
}